// MANN_75539884802609
// MI455X (gfx1250) — compile-verified
//
#include <hip/hip_runtime.h>
#include <math.h>

// ---------------------------------------------------------------------------
// DNC / MANN forward for MI455X (gfx1250, wave32, WMMA).
// Phase 1: G[b*50+t, 0:512] = x[b,t,:789] @ WihX^T + bih + bhh   (big WMMA GEMM)
// Phase 2: per-block 16-batch tile runs all 50 recurrent steps with state in
//          LDS; gates and interface GEMMs use v_wmma_f32_16x16x4_f32.
// All inner K-loops are branch-free (phase-split), so each iteration is just
// two global/ds loads + one v_wmma.
// ---------------------------------------------------------------------------

typedef __attribute__((ext_vector_type(2))) float v2f;
typedef __attribute__((ext_vector_type(8))) float v8f;

__device__ __forceinline__ v8f wmma4(v2f a, v2f b, v8f c) {
  // D = A(16x4,f32) * B(4x16,f32) + C(16x16,f32)
  return __builtin_amdgcn_wmma_f32_16x16x4_f32(
      /*neg_a=*/false, a, /*neg_b=*/false, b,
      /*c_mod=*/(short)0, c, /*reuse_a=*/false, /*reuse_b=*/false);
}

__device__ __forceinline__ float sigm(float x) { return 1.0f / (1.0f + expf(-x)); }
__device__ __forceinline__ float softplusf(float x) {
  return (x > 20.0f) ? x : log1pf(expf(x));
}

// =========================================================================
// Kernel 1: input projection GEMM.
// Rows: 25600 (= B*T). Cols: 512 (= 4*HIDDEN). K: 789 (= 784 img + 5 label).
// Grid: (1600, 4), block 256 (8 waves). Each wave -> one 16x16 C tile.
// A fragment (16x4 f32): lane m = lane&15, K-pair = 2*(lane>>4)+{0,1}.
// =========================================================================
__global__ __launch_bounds__(256) void k_gemm_in(
    const float* __restrict__ imgs,   // (512,10,5,784)
    const float* __restrict__ labs,   // (512,10,5,5)
    const float* __restrict__ Wih,    // (512,853)
    const float* __restrict__ bih,    // (512)
    const float* __restrict__ bhh,    // (512)
    float* __restrict__ G)            // (25600,512)
{
  const int lane = threadIdx.x & 31;
  const int wave = threadIdx.x >> 5;
  const int row0 = blockIdx.x * 16;
  const int col0 = blockIdx.y * 128 + wave * 16;

  const int ho = (lane >> 4) * 2;      // K sub-offset for this half-wave

  // decode global row -> (b, k1, ncls) for the gathered x construction
  const int grow = row0 + (lane & 15); // [0, 25600)
  const int b    = grow / 50;
  const int t    = grow - b * 50;
  const int k1   = t / 5;
  const int ncls = t - k1 * 5;
  const float* arow = imgs + ((long)(b * 10 + k1) * 5 + ncls) * 784;
  const float* lrow = labs + ((long)(b * 10 + k1) * 5 + ncls) * 5;
  const bool lzero = (k1 == 9);        // last presentation: labels zeroed

  const int bcol = col0 + (lane & 15); // B column (output gate index), <512
  const float* wrow = Wih + (long)bcol * 853;

  v8f acc = {};

  // ---- bulk: k in [0, 784), branch-free ----
  #pragma unroll 4
  for (int k = 0; k < 784; k += 4) {
    const int kk = k + ho;
    v2f a, bb;
    a.x  = arow[kk];
    a.y  = arow[kk + 1];
    bb.x = wrow[kk];
    bb.y = wrow[kk + 1];
    acc = wmma4(a, bb, acc);
  }

  // ---- tail step 1: k = 784 (label cols 784..787) ----
  {
    const int kk = 784 + ho;
    v2f a, bb;
    a.x  = lzero ? 0.0f : lrow[kk - 784];
    a.y  = lzero ? 0.0f : lrow[kk - 783];
    bb.x = wrow[kk];
    bb.y = wrow[kk + 1];
    acc = wmma4(a, bb, acc);
  }
  // ---- tail step 2: k = 788 (last label col 788; 789..791 are zero-pad) ----
  {
    const int kk = 788 + ho;            // 788/789 (lanes<16), 790/791 (>=16)
    v2f a, bb;
    a.x  = (ho == 0 && !lzero) ? lrow[4] : 0.0f;
    a.y  = 0.0f;
    bb.x = wrow[kk];                    // <853, safe; multiplied by zero A
    bb.y = wrow[kk + 1];
    acc = wmma4(a, bb, acc);
  }

  const float bias = bih[bcol] + bhh[bcol];
  const int rbase = row0 + (lane >> 4) * 8;   // C: VGPR v -> M = v + 8*(lane>=16)
  #pragma unroll
  for (int v = 0; v < 8; ++v)
    G[(long)(rbase + v) * 512 + bcol] = acc[v] + bias;
}

// =========================================================================
// Kernel 2: recurrent DNC. Grid 32 blocks x 256 threads; block owns 16 batch
// elements for all 50 steps. All state lives in LDS (~93 KB / workgroup).
// =========================================================================
__global__ __launch_bounds__(256) void k_dnc(
    const float* __restrict__ G,      // (25600,512) precomputed input gates
    const float* __restrict__ Wih,    // (512,853)  (rv part = cols 789..852)
    const float* __restrict__ Whh,    // (512,128)
    const float* __restrict__ Witf,   // (264,128)
    const float* __restrict__ bitf,   // (264)
    const float* __restrict__ Wout,   // (5,192)
    const float* __restrict__ bout,   // (5)
    float* __restrict__ out)          // (512,50,5) == (B,K1,N,N) flattened
{
  __shared__ float s_h [16][128];
  __shared__ float s_c [16][128];
  __shared__ float s_rv[16][64];
  __shared__ float s_M [16][5][64];
  __shared__ float s_u [16][5];
  __shared__ float s_ww[16][5];
  __shared__ float s_wr[16][5];
  __shared__ float s_p [16][5];
  __shared__ float s_a [16][5];
  __shared__ float s_cs[16][5];
  __shared__ float s_fw[16][5];
  __shared__ float s_bw[16][5];
  __shared__ float s_L [16][5][5];
  __shared__ float s_g [16][512];
  __shared__ float s_xi[16][272];     // 264 used, padded

  const int tid  = threadIdx.x;
  const int lane = tid & 31;
  const int wave = tid >> 5;
  const int b0   = blockIdx.x * 16;

  // ---- zero initial state ----
  for (int i = tid; i < 16 * 128; i += 256) { (&s_h[0][0])[i] = 0.f; (&s_c[0][0])[i] = 0.f; }
  for (int i = tid; i < 16 * 64;  i += 256) (&s_rv[0][0])[i] = 0.f;
  for (int i = tid; i < 16 * 5 * 64; i += 256) (&s_M[0][0][0])[i] = 0.f;
  for (int i = tid; i < 16 * 5;   i += 256) {
    (&s_u[0][0])[i] = 0.f; (&s_ww[0][0])[i] = 0.f;
    (&s_wr[0][0])[i] = 0.f; (&s_p[0][0])[i] = 0.f;
  }
  for (int i = tid; i < 16 * 25;  i += 256) (&s_L[0][0][0])[i] = 0.f;
  __syncthreads();

  const int m = tid >> 4;   // batch element within tile (DNC stages)
  const int q = tid & 15;   // worker lane within batch element

  for (int t = 0; t < 50; ++t) {
    // ================= gates GEMM: 16x512, K = 192 (h:128 | rv:64) =========
    {
      const int mrow = lane & 15;
      const int ho   = (lane >> 4) * 2;
      const int rb   = (lane >> 4) * 8;
      for (int ct = 0; ct < 4; ++ct) {
        const int col0 = wave * 64 + ct * 16;
        const int bcol = col0 + (lane & 15);
        v8f acc;
        #pragma unroll
        for (int v = 0; v < 8; ++v)
          acc[v] = G[(long)((b0 + rb + v) * 50 + t) * 512 + bcol];

        // phase 1: K in [0,128) -> A = h, B = Whh
        const float* whrow = Whh + (long)bcol * 128;
        #pragma unroll 4
        for (int k = 0; k < 128; k += 4) {
          const int kk = k + ho;
          v2f a, bb;
          a.x  = s_h[mrow][kk];
          a.y  = s_h[mrow][kk + 1];
          bb.x = whrow[kk];
          bb.y = whrow[kk + 1];
          acc = wmma4(a, bb, acc);
        }
        // phase 2: K in [128,192) -> A = rv, B = Wih cols 789..852
        const float* wxrow = Wih + (long)bcol * 853 + 789;
        #pragma unroll 4
        for (int k = 0; k < 64; k += 4) {
          const int kk = k + ho;
          v2f a, bb;
          a.x  = s_rv[mrow][kk];
          a.y  = s_rv[mrow][kk + 1];
          bb.x = wxrow[kk];
          bb.y = wxrow[kk + 1];
          acc = wmma4(a, bb, acc);
        }
        #pragma unroll
        for (int v = 0; v < 8; ++v) s_g[rb + v][bcol] = acc[v];
      }
    }
    __syncthreads();

    // ================= LSTM elementwise =====================================
    for (int i = tid; i < 16 * 128; i += 256) {
      const int mm = i >> 7, j = i & 127;
      const float ig = sigm(s_g[mm][j]);
      const float fg = sigm(s_g[mm][128 + j]);
      const float gg = tanhf(s_g[mm][256 + j]);
      const float og = sigm(s_g[mm][384 + j]);
      const float cn = fg * s_c[mm][j] + ig * gg;
      s_c[mm][j] = cn;
      s_h[mm][j] = og * tanhf(cn);
    }
    __syncthreads();

    // ================= interface GEMM: 16x264, K = 128 ======================
    {
      const int mrow = lane & 15;
      const int ho   = (lane >> 4) * 2;
      const int rb   = (lane >> 4) * 8;
      for (int tile = wave; tile < 17; tile += 8) {
        const int col0 = tile * 16;
        const int bcol = col0 + (lane & 15);
        const bool cv  = (bcol < 264);
        // clamp invalid columns to a safe row: their D columns are discarded
        const float* wrow = Witf + (long)(cv ? bcol : 0) * 128;
        const float bias = cv ? bitf[bcol] : 0.0f;
        v8f acc;
        #pragma unroll
        for (int v = 0; v < 8; ++v) acc[v] = bias;
        #pragma unroll 4
        for (int k = 0; k < 128; k += 4) {
          const int kk = k + ho;
          v2f a, bb;
          a.x  = s_h[mrow][kk];
          a.y  = s_h[mrow][kk + 1];
          bb.x = wrow[kk];
          bb.y = wrow[kk + 1];
          acc = wmma4(a, bb, acc);
        }
        if (cv) {
          #pragma unroll
          for (int v = 0; v < 8; ++v) s_xi[rb + v][bcol] = acc[v];
        }
      }
    }
    __syncthreads();

    // ================= DNC memory update (per batch element m) ==============
    const float* xi = s_xi[m];
    const float br    = 1.0f + softplusf(xi[64]);
    const float bwr   = 1.0f + softplusf(xi[129]);
    const float fgate = sigm(xi[258]);
    const float ga    = sigm(xi[259]);
    const float gwg   = sigm(xi[260]);
    float pa = xi[261], pb = xi[262], pc = xi[263];
    {
      const float pm = fmaxf(pa, fmaxf(pb, pc));
      pa = expf(pa - pm); pb = expf(pb - pm); pc = expf(pc - pm);
      const float ps = pa + pb + pc;
      pa /= ps; pb /= ps; pc /= ps;
    }
    const float pi0 = pa, pi1 = pb, pi2 = pc;

    // usage update (uses ww_prev, wr_prev)
    if (q < 5) {
      const float psi = 1.0f - fgate * s_wr[m][q];
      const float uu = s_u[m][q], wwp = s_ww[m][q];
      s_u[m][q] = (uu + wwp - uu * wwp) * psi;
    }
    __syncthreads();

    // allocation weighting (sort of 5 elements)
    if (q == 0) {
      float u[5]; int idx[5];
      for (int c = 0; c < 5; ++c) { u[c] = s_u[m][c]; idx[c] = c; }
      for (int i = 1; i < 5; ++i) {
        const int key = idx[i]; int j = i - 1;
        while (j >= 0 && u[idx[j]] > u[key]) { idx[j + 1] = idx[j]; --j; }
        idx[j + 1] = key;
      }
      float excl = 1.0f;
      for (int i = 0; i < 5; ++i) {
        const float us = u[idx[i]];
        s_a[m][idx[i]] = (1.0f - us) * excl;
        excl *= us;
      }
    }
    __syncthreads();

    // write content weighting cw (old M, key kw = xi[65..128], beta bwr)
    if (q < 5) {
      float sM = 0.f, sd = 0.f, sk = 0.f;
      for (int w = 0; w < 64; ++w) {
        const float mv = s_M[m][q][w];
        const float kv = xi[65 + w];
        sM += mv * mv; sd += mv * kv; sk += kv * kv;
      }
      const float cosv = sd / ((sqrtf(sM) + 1e-6f) * (sqrtf(sk) + 1e-6f));
      s_cs[m][q] = bwr * cosv;
    }
    __syncthreads();
    if (q == 0) {
      float mx = s_cs[m][0];
      for (int c = 1; c < 5; ++c) mx = fmaxf(mx, s_cs[m][c]);
      float sum = 0.f, ex[5];
      for (int c = 0; c < 5; ++c) { ex[c] = expf(s_cs[m][c] - mx); sum += ex[c]; }
      for (int c = 0; c < 5; ++c) s_cs[m][c] = ex[c] / sum;
    }
    __syncthreads();

    if (q < 5)
      s_ww[m][q] = gwg * (ga * s_a[m][q] + (1.0f - ga) * s_cs[m][q]);
    __syncthreads();

    // memory write: M = M*(1 - ww^T e) + ww^T v
    for (int i = 0; i < 4; ++i) {
      const int w = q + 16 * i;
      const float ev = sigm(xi[130 + w]);
      const float vv = xi[194 + w];
      for (int c = 0; c < 5; ++c) {
        const float wwv = s_ww[m][c];
        s_M[m][c][w] = s_M[m][c][w] * (1.0f - wwv * ev) + wwv * vv;
      }
    }
    __syncthreads();

    // temporal link matrix, then precedence
    if (q < 5) {
      const float wwi = s_ww[m][q];
      for (int j = 0; j < 5; ++j) {
        s_L[m][q][j] = (j == q) ? 0.0f
            : (1.0f - wwi - s_ww[m][j]) * s_L[m][q][j] + wwi * s_p[m][j];
      }
    }
    __syncthreads();
    if (q < 5) {
      float sw = 0.f;
      for (int c = 0; c < 5; ++c) sw += s_ww[m][c];
      s_p[m][q] = (1.0f - sw) * s_p[m][q] + s_ww[m][q];
    }
    __syncthreads();

    // forward / backward read modes (new L, wr_prev)
    if (q < 5) {
      float fv = 0.f, bv = 0.f;
      for (int j = 0; j < 5; ++j) {
        fv += s_L[m][q][j] * s_wr[m][j];
        bv += s_L[m][j][q] * s_wr[m][j];
      }
      s_fw[m][q] = fv; s_bw[m][q] = bv;
    }
    __syncthreads();

    // read content weighting cr (new M, key kr = xi[0..63], beta br)
    if (q < 5) {
      float sM = 0.f, sd = 0.f, sk = 0.f;
      for (int w = 0; w < 64; ++w) {
        const float mv = s_M[m][q][w];
        const float kv = xi[w];
        sM += mv * mv; sd += mv * kv; sk += kv * kv;
      }
      const float cosv = sd / ((sqrtf(sM) + 1e-6f) * (sqrtf(sk) + 1e-6f));
      s_cs[m][q] = br * cosv;
    }
    __syncthreads();
    if (q == 0) {
      float mx = s_cs[m][0];
      for (int c = 1; c < 5; ++c) mx = fmaxf(mx, s_cs[m][c]);
      float sum = 0.f, ex[5];
      for (int c = 0; c < 5; ++c) { ex[c] = expf(s_cs[m][c] - mx); sum += ex[c]; }
      for (int c = 0; c < 5; ++c) s_cs[m][c] = ex[c] / sum;
    }
    __syncthreads();

    if (q < 5)
      s_wr[m][q] = pi0 * s_bw[m][q] + pi1 * s_cs[m][q] + pi2 * s_fw[m][q];
    __syncthreads();

    // read vector rv = wr @ M_new
    for (int i = 0; i < 4; ++i) {
      const int w = q + 16 * i;
      float r = 0.f;
      for (int c = 0; c < 5; ++c) r += s_wr[m][c] * s_M[m][c][w];
      s_rv[m][w] = r;
    }
    __syncthreads();

    // output projection y = [h, rv] @ Wout^T + bout  (16x5, K=192)
    if (tid < 80) {
      const int mm = tid / 5, n = tid % 5;
      const float* wr_ = Wout + n * 192;
      float acc = bout[n];
      for (int k = 0; k < 128; ++k) acc += wr_[k] * s_h[mm][k];
      for (int k = 0; k < 64;  ++k) acc += wr_[128 + k] * s_rv[mm][k];
      out[(long)((b0 + mm) * 50 + t) * 5 + n] = acc;
    }
    __syncthreads();
  }
}

// =========================================================================
// Launch
// =========================================================================
extern "C" void kernel_launch(void* const* d_in, const int* in_sizes, int n_in,
                              void* d_out, int out_size, void* d_ws, size_t ws_size,
                              hipStream_t stream) {
  const float* imgs = (const float*)d_in[0];   // (512,10,5,784)
  const float* labs = (const float*)d_in[1];   // (512,10,5,5)
  const float* Wih  = (const float*)d_in[2];   // (512,853)
  const float* Whh  = (const float*)d_in[3];   // (512,128)
  const float* bih  = (const float*)d_in[4];   // (512)
  const float* bhh  = (const float*)d_in[5];   // (512)
  const float* Witf = (const float*)d_in[6];   // (264,128)
  const float* bitf = (const float*)d_in[7];   // (264)
  const float* Wout = (const float*)d_in[8];   // (5,192)
  const float* bout = (const float*)d_in[9];   // (5)
  float* outp = (float*)d_out;                 // (512,10,5,5)
  float* G = (float*)d_ws;                     // 25600*512 floats = 52.4 MB

  (void)in_sizes; (void)n_in; (void)out_size; (void)ws_size;

  k_gemm_in<<<dim3(1600, 4), 256, 0, stream>>>(imgs, labs, Wih, bih, bhh, G);
  k_dnc<<<32, 256, 0, stream>>>(G, Wih, Whh, Witf, bitf, Wout, bout, outp);
}